// GlobalPool_55568286876341
// MI455X (gfx1250) — compile-verified
//
#include <hip/hip_runtime.h>
#include <math.h>

#define FEAT 256
#define NNODES 262144
#define NGRAPHS 8192
#define LEAKY 0.01f

#if defined(__HIP_DEVICE_COMPILE__)
#if !__has_builtin(__builtin_amdgcn_wmma_f32_16x16x4_f32)
#error "missing __builtin_amdgcn_wmma_f32_16x16x4_f32"
#endif
#endif

typedef float v2f __attribute__((ext_vector_type(2)));
typedef float v8f __attribute__((ext_vector_type(8)));

__device__ __forceinline__ float wave_sum(float v) {
  for (int o = 16; o > 0; o >>= 1) v += __shfl_xor(v, o, 32);
  return v;
}

// ---- Kernel 1: gfa[g] = sum_k relu(g_feats[g,k]) * W_log[k]  (k in [0,256))
__global__ void graph_logit_dot(const float* __restrict__ gf,
                                const float* __restrict__ Wlog,
                                float* __restrict__ gfa) {
  __shared__ float red[256];
  int g = blockIdx.x, t = threadIdx.x;
  float x = gf[(size_t)g * FEAT + t];
  x = x > 0.f ? x : 0.f;
  red[t] = x * Wlog[t];
  __syncthreads();
  for (int s = 128; s > 0; s >>= 1) {
    if (t < s) red[t] += red[t + s];
    __syncthreads();
  }
  if (t == 0) gfa[g] = red[0];
}

// ---- Kernel 2: z[i] = leakyrelu(nf[i,:]·W_log[256:512] + gfa[seg[i]] + b_log)
__global__ void node_logits(const float* __restrict__ nf,
                            const float* __restrict__ Wlog,
                            const float* __restrict__ blog,
                            const float* __restrict__ gfa,
                            const int* __restrict__ seg,
                            float* __restrict__ z) {
  int wave = (int)((blockIdx.x * (size_t)blockDim.x + threadIdx.x) >> 5);
  int lane = threadIdx.x & 31;
  if (wave >= NNODES) return;
  const float* row = nf + (size_t)wave * FEAT + lane * 8;
  const float* w = Wlog + FEAT + lane * 8;
  float s = 0.f;
#pragma unroll
  for (int j = 0; j < 8; ++j) s += row[j] * w[j];
  s = wave_sum(s);
  if (lane == 0) {
    float v = s + gfa[seg[wave]] + blog[0];
    z[wave] = v >= 0.f ? v : LEAKY * v;
  }
}

// ---- Kernel 3: segment start offsets from sorted seg -> starts[0..NGRAPHS]
__global__ void seg_starts(const int* __restrict__ seg, int* __restrict__ starts) {
  int i = blockIdx.x * blockDim.x + threadIdx.x;
  if (i >= NNODES) return;
  int s = seg[i];
  if (i == 0) {
    for (int g = 0; g <= s; ++g) starts[g] = 0;
  } else {
    int p = seg[i - 1];
    for (int g = p + 1; g <= s; ++g) starts[g] = i;
  }
  if (i == NNODES - 1) {
    for (int g = s + 1; g <= NGRAPHS; ++g) starts[g] = NNODES;
  }
}

// ---- Kernel 4: per-graph softmax over z, then y[g,f] = sum_i a_i * nf[i,f]
__global__ void seg_softmax_wsum(const float* __restrict__ nf,
                                 const float* __restrict__ z,
                                 const int* __restrict__ starts,
                                 float* __restrict__ y) {
  __shared__ float red[256];
  __shared__ float wbuf[256];
  __shared__ float sh_m, sh_s;
  int g = blockIdx.x, t = threadIdx.x;
  int s0 = starts[g], e0 = starts[g + 1];
  if (s0 >= e0) {  // empty graph -> zero contribution (matches reference)
    y[(size_t)g * FEAT + t] = 0.f;
    return;
  }
  float lm = -INFINITY;
  for (int i = s0 + t; i < e0; i += 256) lm = fmaxf(lm, z[i]);
  red[t] = lm;
  __syncthreads();
  for (int s = 128; s > 0; s >>= 1) {
    if (t < s) red[t] = fmaxf(red[t], red[t + s]);
    __syncthreads();
  }
  if (t == 0) sh_m = red[0];
  __syncthreads();
  float m = sh_m;
  float ls = 0.f;
  for (int i = s0 + t; i < e0; i += 256) ls += expf(z[i] - m);
  red[t] = ls;
  __syncthreads();
  for (int s = 128; s > 0; s >>= 1) {
    if (t < s) red[t] += red[t + s];
    __syncthreads();
  }
  if (t == 0) sh_s = red[0];
  __syncthreads();
  float invs = 1.f / sh_s;
  float acc = 0.f;
  for (int c0 = s0; c0 < e0; c0 += 256) {
    int cn = min(256, e0 - c0);
    __syncthreads();
    if (t < cn) wbuf[t] = expf(z[c0 + t] - m);
    __syncthreads();
    for (int i = 0; i < cn; ++i) acc += wbuf[i] * nf[(size_t)(c0 + i) * FEAT + t];
  }
  y[(size_t)g * FEAT + t] = acc * invs;
}

// ---- WMMA f32 16x16x4 GEMM: C = act(A[MxK] * B + bias)
// BT=0: B is [K x N] row-major (B[k*ldb+n]); BT=1: B is [N x K] row-major (B[n*ldb+k], i.e. A@B^T)
// ACT=1: elu.  RMASK=1: bias zeroed for rows whose segment is empty.
template <int BT, int ACT, int RMASK>
__global__ void gemm_wmma_f32(const float* __restrict__ A, int lda,
                              const float* __restrict__ Bm, int ldb,
                              const float* __restrict__ bias,
                              const int* __restrict__ starts,
                              float* __restrict__ C, int ldc,
                              int M, int N, int K) {
  int tile = blockIdx.x * (blockDim.x >> 5) + (threadIdx.x >> 5);
  int colTiles = N >> 4;
  int rowT = tile / colTiles;
  int colT = tile - rowT * colTiles;
  if (rowT >= (M >> 4)) return;  // wave-uniform: EXEC stays all-ones for WMMA
  int lane = threadIdx.x & 31;
  int half = lane >> 4;   // selects K pair {0,1} vs {2,3} within each 4-step
  int l16 = lane & 15;
  int arow = (rowT << 4) + l16;  // A fragment row for this lane
  int col = (colT << 4) + l16;   // B/C/D column for this lane
  v8f acc = {0.f, 0.f, 0.f, 0.f, 0.f, 0.f, 0.f, 0.f};
  const float* aPtr = A + (size_t)arow * lda + half * 2;
  for (int k0 = 0; k0 < K; k0 += 4) {
    v2f a, b;
    a.x = aPtr[k0];
    a.y = aPtr[k0 + 1];
    int kb = k0 + half * 2;
    if (BT) {
      const float* bp = Bm + (size_t)col * ldb + kb;
      b.x = bp[0];
      b.y = bp[1];
    } else {
      b.x = Bm[(size_t)kb * ldb + col];
      b.y = Bm[(size_t)(kb + 1) * ldb + col];
    }
    acc = __builtin_amdgcn_wmma_f32_16x16x4_f32(false, a, false, b, (short)0,
                                                acc, false, false);
  }
  // D layout: VGPR r holds row (half*8 + r), column = lane&15
  int rowBase = (rowT << 4) + half * 8;
#pragma unroll
  for (int r = 0; r < 8; ++r) {
    int rr = rowBase + r;
    float bv = bias[col];
    if (RMASK) {
      if (starts[rr] >= starts[rr + 1]) bv = 0.f;  // empty graph -> g_repr exactly 0
    }
    float v = acc[r] + bv;
    if (ACT == 1) v = v > 0.f ? v : expm1f(v);  // elu
    C[(size_t)rr * ldc + col] = v;
  }
}

// ---- Kernel 7: GRU cell elementwise
__global__ void gru_elem(const float* __restrict__ gi, const float* __restrict__ gh,
                         const float* __restrict__ gf, float* __restrict__ out) {
  int idx = blockIdx.x * blockDim.x + threadIdx.x;
  if (idx >= NGRAPHS * FEAT) return;
  int b = idx >> 8, f = idx & 255;
  const float* gib = gi + (size_t)b * 768;
  const float* ghb = gh + (size_t)b * 768;
  float ir = gib[f], iz = gib[256 + f], in = gib[512 + f];
  float hr = ghb[f], hz = ghb[256 + f], hn = ghb[512 + f];
  float r = 1.f / (1.f + expf(-(ir + hr)));
  float zz = 1.f / (1.f + expf(-(iz + hz)));
  float n = tanhf(in + r * hn);
  float h = gf[idx];
  out[idx] = (1.f - zz) * n + zz * h;
}

// ---- workspace layout (bytes)
static const size_t OFS_GFA = 0;                              // 8192 f32
static const size_t OFS_Z = 32768;                            // 262144 f32
static const size_t OFS_STARTS = OFS_Z + (size_t)NNODES * 4;  // 8193 i32
static const size_t OFS_Y = ((OFS_STARTS + (size_t)(NGRAPHS + 1) * 4 + 255) / 256) * 256;
static const size_t OFS_CTX = OFS_Y + (size_t)NGRAPHS * FEAT * 4;
static const size_t OFS_GI = OFS_CTX + (size_t)NGRAPHS * FEAT * 4;
static const size_t OFS_GH = OFS_GI + (size_t)NGRAPHS * 768 * 4;

extern "C" void kernel_launch(void* const* d_in, const int* in_sizes, int n_in,
                              void* d_out, int out_size, void* d_ws, size_t ws_size,
                              hipStream_t stream) {
  const float* nf = (const float*)d_in[0];    // [N, 256]
  const float* gf = (const float*)d_in[1];    // [B, 256]
  const int* seg = (const int*)d_in[2];       // [N]
  const float* Wlog = (const float*)d_in[3];  // [512, 1]
  const float* blog = (const float*)d_in[4];  // [1]
  const float* Wproj = (const float*)d_in[5]; // [256, 256]
  const float* bproj = (const float*)d_in[6]; // [256]
  const float* Wih = (const float*)d_in[7];   // [768, 256]
  const float* Whh = (const float*)d_in[8];   // [768, 256]
  const float* bih = (const float*)d_in[9];   // [768]
  const float* bhh = (const float*)d_in[10];  // [768]
  float* out = (float*)d_out;

  uint8_t* ws = (uint8_t*)d_ws;
  float* gfa = (float*)(ws + OFS_GFA);
  float* z = (float*)(ws + OFS_Z);
  int* starts = (int*)(ws + OFS_STARTS);
  float* y = (float*)(ws + OFS_Y);
  float* ctx = (float*)(ws + OFS_CTX);
  float* gi = (float*)(ws + OFS_GI);
  float* gh = (float*)(ws + OFS_GH);

  graph_logit_dot<<<NGRAPHS, 256, 0, stream>>>(gf, Wlog, gfa);
  node_logits<<<NNODES / 8, 256, 0, stream>>>(nf, Wlog, blog, gfa, seg, z);
  seg_starts<<<NNODES / 256, 256, 0, stream>>>(seg, starts);
  seg_softmax_wsum<<<NGRAPHS, 256, 0, stream>>>(nf, z, starts, y);

  // context = elu(y @ W_proj + b_proj), bias masked for empty graphs
  {
    int tiles = (NGRAPHS / 16) * (FEAT / 16);  // 8192
    gemm_wmma_f32<0, 1, 1><<<tiles / 4, 128, 0, stream>>>(
        y, FEAT, Wproj, FEAT, bproj, starts, ctx, FEAT, NGRAPHS, FEAT, FEAT);
  }
  // gi = ctx @ W_ih^T + b_ih ; gh = gf @ W_hh^T + b_hh
  {
    int tiles = (NGRAPHS / 16) * (768 / 16);  // 24576
    gemm_wmma_f32<1, 0, 0><<<tiles / 4, 128, 0, stream>>>(
        ctx, FEAT, Wih, FEAT, bih, nullptr, gi, 768, NGRAPHS, 768, FEAT);
    gemm_wmma_f32<1, 0, 0><<<tiles / 4, 128, 0, stream>>>(
        gf, FEAT, Whh, FEAT, bhh, nullptr, gh, 768, NGRAPHS, 768, FEAT);
  }
  gru_elem<<<(NGRAPHS * FEAT) / 256, 256, 0, stream>>>(gi, gh, gf, out);
}